// Test_Model_26388279066907
// MI455X (gfx1250) — compile-verified
//
#include <hip/hip_runtime.h>
#include <math.h>

// Channel-diagonal Clebsch-Gordan product (MAXL=2), B=1024, T=512.
// Stage 1 (VALU): per GEMM column (channel x {re,im}) compute pair products
//   t[pq] = P[gp]*s0[gq] + Q[gp]*s1[gq]   (re: rr-ii / im: ri+ir via s0/s1 select)
// Stage 2 (WMMA f32 16x16x4): out[81 rows] = W[81x81 block-diag CG] @ t, with
//   only the k-steps intersecting each 16-row M-tile's nonzero blocks (35 WMMAs).
// W built per call by init kernel (Racah formula, f64) into d_ws; staged in LDS.

#define T_DIM    512
#define B_DIM    1024
#define W_STRIDE 84
#define W_ROWS   80

typedef float v8f __attribute__((ext_vector_type(8)));
typedef float v2f __attribute__((ext_vector_type(2)));

// pq -> global component index (0 = l0; 1..3 = l1; 4..8 = l2), pair order:
// (2,2)[0..24] (1,2)[25..39] (2,1)[40..54] (1,1)[55..63] (0,2)[64..68]
// (2,0)[69..73] (0,1)[74..76] (1,0)[77..79] (0,0)[80]
__device__ constexpr int PQ_GP[81] = {
  4,4,4,4,4, 5,5,5,5,5, 6,6,6,6,6, 7,7,7,7,7, 8,8,8,8,8,
  1,1,1,1,1, 2,2,2,2,2, 3,3,3,3,3,
  4,4,4, 5,5,5, 6,6,6, 7,7,7, 8,8,8,
  1,1,1, 2,2,2, 3,3,3,
  0,0,0,0,0,
  4,5,6,7,8,
  0,0,0,
  1,2,3,
  0};
__device__ constexpr int PQ_GQ[81] = {
  4,5,6,7,8, 4,5,6,7,8, 4,5,6,7,8, 4,5,6,7,8, 4,5,6,7,8,
  4,5,6,7,8, 4,5,6,7,8, 4,5,6,7,8,
  1,2,3, 1,2,3, 1,2,3, 1,2,3, 1,2,3,
  1,2,3, 1,2,3, 1,2,3,
  4,5,6,7,8,
  0,0,0,0,0,
  1,2,3,
  0,0,0,
  0};

// output row -> (l, m+l, segment) ; rows grouped identically to pq pair blocks
__device__ constexpr int ROW_L[81] = {
  0, 1,1,1, 2,2,2,2,2, 3,3,3,3,3,3,3, 4,4,4,4,4,4,4,4,4,
  1,1,1, 2,2,2,2,2, 3,3,3,3,3,3,3,
  1,1,1, 2,2,2,2,2, 3,3,3,3,3,3,3,
  0, 1,1,1, 2,2,2,2,2,
  2,2,2,2,2,
  2,2,2,2,2,
  1,1,1,
  1,1,1,
  0};
__device__ constexpr int ROW_MM[81] = {
  0, 0,1,2, 0,1,2,3,4, 0,1,2,3,4,5,6, 0,1,2,3,4,5,6,7,8,
  0,1,2, 0,1,2,3,4, 0,1,2,3,4,5,6,
  0,1,2, 0,1,2,3,4, 0,1,2,3,4,5,6,
  0, 0,1,2, 0,1,2,3,4,
  0,1,2,3,4,
  0,1,2,3,4,
  0,1,2,
  0,1,2,
  0};
__device__ constexpr int ROW_SEG[81] = {
  2, 5,5,5, 5,5,5,5,5, 2,2,2,2,2,2,2, 0,0,0,0,0,0,0,0,0,
  3,3,3, 2,2,2,2,2, 0,0,0,0,0,0,0,
  4,4,4, 4,4,4,4,4, 1,1,1,1,1,1,1,
  1, 2,2,2, 1,1,1,1,1,
  0,0,0,0,0,
  3,3,3,3,3,
  0,0,0,
  1,1,1,
  0};
__device__ constexpr int      NSEG[5]  = {3,6,6,3,1};
__device__ constexpr unsigned OBASE[5] = {0u, 3145728u, 22020096u, 53477376u, 75497472u};
// k-step (4 cols each) ranges touching each 16-row M-tile of block-diagonal W
__device__ constexpr int KSLO[5] = {0,0,6,10,16};
__device__ constexpr int KSHI[5] = {6,9,13,15,19};

// ---------------- W init (Racah formula) ----------------
__device__ constexpr int PAIR_OFF[10] = {0,25,40,55,64,69,74,77,80,81};
__device__ constexpr int PAIR_L1[9]   = {2,1,2,1,0,2,0,1,0};
__device__ constexpr int PAIR_L2[9]   = {2,2,1,1,2,0,1,0,0};

__device__ inline double dfact(int n) {
  double r = 1.0;
  for (int i = 2; i <= n; ++i) r *= (double)i;
  return r;
}

__device__ double clebsch_d(int j1, int m1, int j2, int m2, int j, int m) {
  if (m1 + m2 != m) return 0.0;
  double pre = sqrt((2.0 * j + 1.0) * dfact(j + j1 - j2) * dfact(j - j1 + j2) *
                    dfact(j1 + j2 - j) / dfact(j1 + j2 + j + 1));
  pre *= sqrt(dfact(j + m) * dfact(j - m) * dfact(j1 - m1) * dfact(j1 + m1) *
              dfact(j2 - m2) * dfact(j2 + m2));
  int vmin = 0;
  if (j2 - j - m1 > vmin) vmin = j2 - j - m1;
  if (j1 + m2 - j > vmin) vmin = j1 + m2 - j;
  int vmax = j1 + j2 - j;
  if (j1 - m1 < vmax) vmax = j1 - m1;
  if (j2 + m2 < vmax) vmax = j2 + m2;
  double s = 0.0;
  for (int v = vmin; v <= vmax; ++v) {
    double d = dfact(v) * dfact(j1 + j2 - j - v) * dfact(j1 - m1 - v) *
               dfact(j2 + m2 - v) * dfact(j - j2 + m1 + v) * dfact(j - j1 - m2 + v);
    s += ((v & 1) ? -1.0 : 1.0) / d;
  }
  return pre * s;
}

__global__ void init_w_kernel(float* __restrict__ W) {
  int idx = blockIdx.x * 256 + threadIdx.x;
  if (idx >= W_ROWS * W_STRIDE) return;
  int row = idx / W_STRIDE, k = idx % W_STRIDE;
  float v = 0.0f;
  if (k < 81) {
    int pr = 0; while (pr < 8 && row >= PAIR_OFF[pr + 1]) ++pr;
    int pk = 0; while (pk < 8 && k   >= PAIR_OFF[pk + 1]) ++pk;
    if (pr == pk) {
      int l1 = PAIR_L1[pr], l2 = PAIR_L2[pr];
      int rin = row - PAIR_OFF[pr];
      int l = (l1 > l2) ? (l1 - l2) : (l2 - l1);
      while (rin >= 2 * l + 1) { rin -= 2 * l + 1; ++l; }
      int mm = rin;
      int i2 = k - PAIR_OFF[pk];
      int s2 = 2 * l2 + 1;
      int p = i2 / s2, q = i2 % s2;
      int m1 = p - l1, m2 = q - l2, m = mm - l;
      if (m1 + m2 == m) v = (float)clebsch_d(l1, m1, l2, m2, l, m);
    }
  }
  W[idx] = v;
}

// ---------------- main kernel ----------------
#define TVAL(kk) (P[PQ_GP[(kk)]] * s0[PQ_GQ[(kk)]] + Q[PQ_GP[(kk)]] * s1[PQ_GQ[(kk)]])

__global__ __launch_bounds__(256) void cg_wmma_kernel(
    const float* __restrict__ x0, const float* __restrict__ x1,
    const float* __restrict__ x2, float* __restrict__ out,
    const float* __restrict__ W) {
  __shared__ float Wl[W_ROWS * W_STRIDE];  // 26.9 KB, stride-84 -> conflict-free b64
  for (int i = threadIdx.x; i < W_ROWS * W_STRIDE; i += 256) Wl[i] = W[i];
  __syncthreads();

  const int  lane = threadIdx.x & 31;
  const int  wv   = threadIdx.x >> 5;
  const int  col  = lane & 15;       // GEMM column within tile = channel*2 + comp
  const bool hi   = lane >= 16;      // K-half per 32-bit WMMA operand layout
  const int  comp = col & 1;

  const int chbase = (blockIdx.x * 8 + wv) * 8;  // 8 channels per wave, same b
  const int b  = chbase >> 9;                    // T = 512
  const int c0 = chbase & (T_DIM - 1);
  const int c  = c0 + (col >> 1);

  const float2* x0v = (const float2*)x0;
  const float2* x1v = (const float2*)x1;
  const float2* x2v = (const float2*)x2;

  float P[9], Q[9];
  {
    float2 e = x0v[b * T_DIM + c]; P[0] = e.x; Q[0] = e.y;
#pragma unroll
    for (int p = 0; p < 3; ++p) {
      float2 e1 = x1v[(b * 3 + p) * T_DIM + c]; P[1 + p] = e1.x; Q[1 + p] = e1.y;
    }
#pragma unroll
    for (int q = 0; q < 5; ++q) {
      float2 e2 = x2v[(b * 5 + q) * T_DIM + c]; P[4 + q] = e2.x; Q[4 + q] = e2.y;
    }
  }
  // comp==0: t = Pr*Pr' - Qi*Qi' (real) ; comp==1: t = Pr*Qi' + Qi*Pr' (imag)
  float s0[9], s1[9];
#pragma unroll
  for (int j = 0; j < 9; ++j) {
    s0[j] = comp ? Q[j] : P[j];
    s1[j] = comp ? P[j] : -Q[j];
  }

  // B-operand values: VGPR0 = t[4ks + (hi?2:0)], VGPR1 = t[4ks+1 + (hi?2:0)]
  float tA[20], tB[20];
#pragma unroll
  for (int ks = 0; ks < 20; ++ks) {
    const float t0 = TVAL(4 * ks + 0);
    const float t1 = TVAL(4 * ks + 1);
    const float t2 = TVAL(4 * ks + 2);
    const float t3 = TVAL(4 * ks + 3);
    tA[ks] = hi ? t2 : t0;
    tB[ks] = hi ? t3 : t1;
  }

  const unsigned laneoff = (unsigned)(c0 * 2 + col);  // == (c*2 + comp) - ...

#pragma unroll
  for (int mt = 0; mt < 5; ++mt) {
    v8f acc = {};
#pragma unroll
    for (int ks = KSLO[mt]; ks <= KSHI[mt]; ++ks) {
      // A operand: W[16mt + (lane&15)][4ks + (hi?2:0) .. +1]  (16x4 f32 layout)
      const float* ap = &Wl[(16 * mt + col) * W_STRIDE + 4 * ks + (hi ? 2 : 0)];
      v2f a = *(const v2f*)ap;
      v2f bb = {tA[ks], tB[ks]};
      acc = __builtin_amdgcn_wmma_f32_16x16x4_f32(false, a, false, bb,
                                                  (short)0, acc, false, false);
    }
    // D: VGPR v -> row 16mt+v (lanes 0-15) / 16mt+v+8 (lanes 16-31), N = lane&15
#pragma unroll
    for (int v = 0; v < 8; ++v) {
      const int rA = 16 * mt + v, rB = rA + 8;
      const unsigned cA = OBASE[ROW_L[rA]] +
          (unsigned)((ROW_MM[rA] * NSEG[ROW_L[rA]] + ROW_SEG[rA]) * (T_DIM * 2));
      const unsigned dA =
          (unsigned)((2 * ROW_L[rA] + 1) * NSEG[ROW_L[rA]] * (T_DIM * 2));
      const unsigned cB = OBASE[ROW_L[rB]] +
          (unsigned)((ROW_MM[rB] * NSEG[ROW_L[rB]] + ROW_SEG[rB]) * (T_DIM * 2));
      const unsigned dB =
          (unsigned)((2 * ROW_L[rB] + 1) * NSEG[ROW_L[rB]] * (T_DIM * 2));
      const unsigned offA = cA + (unsigned)b * dA;
      const unsigned offB = cB + (unsigned)b * dB;
      out[(hi ? offB : offA) + laneoff] = acc[v];
    }
  }

  // Row 80 = (0,0) pair, CG == 1: out0 segment 0, done directly on VALU.
  {
    const float t80 = P[0] * s0[0] + Q[0] * s1[0];
    if (!hi) out[(unsigned)b * 3072u + laneoff] = t80;
  }
}

extern "C" void kernel_launch(void* const* d_in, const int* in_sizes, int n_in,
                              void* d_out, int out_size, void* d_ws, size_t ws_size,
                              hipStream_t stream) {
  (void)in_sizes; (void)n_in; (void)out_size; (void)ws_size;
  const float* x0 = (const float*)d_in[0];
  const float* x1 = (const float*)d_in[1];
  const float* x2 = (const float*)d_in[2];
  float* out = (float*)d_out;
  float* W   = (float*)d_ws;  // 80*84*4 = 26,880 bytes of scratch

  init_w_kernel<<<(W_ROWS * W_STRIDE + 255) / 256, 256, 0, stream>>>(W);
  cg_wmma_kernel<<<(B_DIM * T_DIM) / 64, 256, 0, stream>>>(x0, x1, x2, out, W);
}